// EdgewiseEnergySum_71365176590609
// MI455X (gfx1250) — compile-verified
//
#include <hip/hip_runtime.h>
#include <hip/hip_bf16.h>

// ---------------------------------------------------------------------------
// EdgewiseEnergySum for MI455X (gfx1250, wave32)
//
// Memory-bound scatter:
//   - stream 16M edges (f32 length + 2x int32 index) ~= 200 MB -> ~9us floor
//     at 23.3 TB/s HBM; everything else (species 2MB, out 2MB, scales 100B)
//     lives in L2/WGP$.
//   - segment_sum via hardware global_atomic_add_f32 (no-return, STOREcnt
//     path); uniform random centers -> ~32 adds/atom, no hotspot.
// WMMA/TDM not applicable: no matrix structure, no staged reuse. Optimal
// CDNA5 shape = b128 vector streams + global_prefetch + f32 L2 atomics.
// ---------------------------------------------------------------------------

#define R_MAX_INV (1.0f / 6.0f)

__device__ __forceinline__ float edge_energy(float r, float l0) {
    // poly_cutoff, p=6:  1 - 28 x^6 + 48 x^7 - 21 x^8,  x = r/6, zero for x>=1
    float x  = r * R_MAX_INV;
    float x2 = x * x;
    float x4 = x2 * x2;
    float x6 = x4 * x2;
    float x7 = x6 * x;
    float x8 = x7 * x;
    float cut = 1.0f + fmaf(-28.0f, x6, fmaf(48.0f, x7, -21.0f * x8));
    cut = (x < 1.0f) ? cut : 0.0f;

    // (r/l0)^-12 / 24 * l0  ==  (l0/r)^12 * l0 / 24
    float inv   = l0 / r;
    float inv2  = inv * inv;
    float inv4  = inv2 * inv2;
    float inv12 = inv4 * inv4 * inv4;
    return inv12 * (l0 * (1.0f / 24.0f)) * cut;
}

__global__ void __launch_bounds__(256)
edge_scatter_kernel(const float* __restrict__ len,
                    const int*   __restrict__ centers,
                    const int*   __restrict__ neighbors,
                    const int*   __restrict__ species,
                    const float* __restrict__ scales,
                    float*       __restrict__ out,
                    int E4,      // number of full 4-edge groups
                    int T) {
    // per_edge_scales (T*T <= 64 floats) staged in LDS: divergent (c,n)
    // lookups hit LDS banks instead of extra VMEM gathers.
    __shared__ float s_scales[64];
    if (threadIdx.x < (unsigned)(T * T)) s_scales[threadIdx.x] = scales[threadIdx.x];
    __syncthreads();

    int g = blockIdx.x * blockDim.x + threadIdx.x;
    if (g >= E4) return;
    int i0 = g * 4;

    // prefetch the streaming arrays ~16KB ahead (global_prefetch_b8)
    const int PF = 4096;
    if (g + 1024 < E4) {
        __builtin_prefetch(len       + i0 + PF, 0, 0);
        __builtin_prefetch(centers   + i0 + PF, 0, 0);
        __builtin_prefetch(neighbors + i0 + PF, 0, 0);
    }

    // three b128 vector streams: 4 edges per thread
    float4 r4 = *(const float4*)(len       + i0);
    int4   c4 = *(const int4*)  (centers   + i0);
    int4   n4 = *(const int4*)  (neighbors + i0);

    float r[4]  = {r4.x, r4.y, r4.z, r4.w};
    int   c[4]  = {c4.x, c4.y, c4.z, c4.w};
    int   nb[4] = {n4.x, n4.y, n4.z, n4.w};

    #pragma unroll
    for (int j = 0; j < 4; ++j) {
        int   sc = species[c[j]];
        int   sn = species[nb[j]];
        float l0 = s_scales[sc * T + sn];
        float e  = edge_energy(r[j], l0);
        unsafeAtomicAdd(out + c[j], e);   // global_atomic_add_f32, no return
    }
}

// Scalar cleanup for E % 4 leftover edges (at most 3; one tiny block).
__global__ void edge_tail_kernel(const float* __restrict__ len,
                                 const int*   __restrict__ centers,
                                 const int*   __restrict__ neighbors,
                                 const int*   __restrict__ species,
                                 const float* __restrict__ scales,
                                 float*       __restrict__ out,
                                 int start, int E, int T) {
    int i = start + blockIdx.x * blockDim.x + threadIdx.x;
    if (i >= E) return;
    int   cc = centers[i];
    int   sc = species[cc];
    int   sn = species[neighbors[i]];
    float l0 = scales[sc * T + sn];
    float e  = edge_energy(len[i], l0);
    unsafeAtomicAdd(out + cc, e);
}

// out[i] = per_atom_energy[i]; rewritten every launch so graph replays are
// deterministic (atomics accumulate on top of this).
__global__ void init_out_kernel(const float* __restrict__ pae,
                                float* __restrict__ out, int N4, int N) {
    int g = blockIdx.x * blockDim.x + threadIdx.x;
    if (g < N4) {
        int i0 = g * 4;
        *(float4*)(out + i0) = *(const float4*)(pae + i0);
    } else if (g == N4) {
        for (int i = N4 * 4; i < N; ++i) out[i] = pae[i];
    }
}

extern "C" void kernel_launch(void* const* d_in, const int* in_sizes, int n_in,
                              void* d_out, int out_size, void* d_ws, size_t ws_size,
                              hipStream_t stream) {
    // setup_inputs() order:
    //   0: edge_length      (E,)   f32
    //   1: edge_index       (2,E)  int
    //   2: species          (N,1)  int
    //   3: per_atom_energy  (N,1)  f32
    //   4: per_edge_scales  (T,T)  f32
    const float* edge_length     = (const float*)d_in[0];
    const int*   edge_index      = (const int*)  d_in[1];
    const int*   species         = (const int*)  d_in[2];
    const float* per_atom_energy = (const float*)d_in[3];
    const float* per_edge_scales = (const float*)d_in[4];
    float*       out             = (float*)d_out;

    const int E  = in_sizes[0];
    const int N  = in_sizes[2];
    const int TT = in_sizes[4];
    int T = 1;
    while (T * T < TT) ++T;   // T = 5 for the reference setup

    const int* centers   = edge_index;       // row 0
    const int* neighbors = edge_index + E;   // row 1

    // 1) out = per_atom_energy
    {
        int N4 = N / 4;
        int threads = N4 + 1;                // +1 thread handles the tail
        int blocks  = (threads + 255) / 256;
        init_out_kernel<<<blocks, 256, 0, stream>>>(per_atom_energy, out, N4, N);
    }
    // 2) accumulate edge energies into out
    {
        int E4 = E / 4;
        int blocks = (E4 + 255) / 256;       // 16384 blocks for E = 16M
        edge_scatter_kernel<<<blocks, 256, 0, stream>>>(
            edge_length, centers, neighbors, species, per_edge_scales, out, E4, T);
        int rem = E - E4 * 4;
        if (rem > 0) {
            edge_tail_kernel<<<1, 32, 0, stream>>>(
                edge_length, centers, neighbors, species, per_edge_scales, out,
                E4 * 4, E, T);
        }
    }
}